// GaussianBlur_61426622267936
// MI455X (gfx1250) — compile-verified
//
#include <hip/hip_runtime.h>

typedef float v2f __attribute__((ext_vector_type(2)));
typedef float v8f __attribute__((ext_vector_type(8)));

#define IMG_H 512
#define IMG_W 512
#define BLK_W 128           // columns per block (8 wave strips of 16)
#define BLK_H 64            // rows per block (4 WMMA row-tiles per wave)
#define XS_ROWS 66          // BLK_H + 2 halo rows
#define XS_STRIDE 140       // floats; 140 mod 64 = 12 -> 16-row column reads are
                            // bank-conflict-free; interior base col 4 is 16B aligned
// LDS row layout: [3]=left halo (col0-1), [4..131]=col0..col0+127, [132]=right halo (col0+128)

__device__ __forceinline__ int reflect101(int i, int n) {
    i = (i < 0) ? -i : i;
    i = (i >= n) ? (2 * n - 2 - i) : i;
    return i;
}

__global__ __launch_bounds__(256)
void gauss3x3_wmma_f32(const float* __restrict__ x, float* __restrict__ out) {
    __shared__ float xs[XS_ROWS * XS_STRIDE];   // 36,960 B

    const int img  = blockIdx.z;                // b*3 + c, 0..191
    const int row0 = blockIdx.y * BLK_H;        // 0..448
    const int col0 = blockIdx.x * BLK_W;        // 0..384
    const float* __restrict__ src = x   + (size_t)img * (IMG_H * IMG_W);
    float* __restrict__       dst = out + (size_t)img * (IMG_H * IMG_W);

    const int tid = threadIdx.x;

    // ---- Stage interior: each thread owns one float4 column slot, marches rows.
    // Wave = one full 128-col row per iteration -> perfectly coalesced 512B b128 loads.
    {
        const int q = tid & 31;
        const float* gsrc = src + col0 + 4 * q;
        float* lcol = &xs[4 + 4 * q];
        for (int r = tid >> 5; r < XS_ROWS; r += 8) {
            int gr = reflect101(row0 - 1 + r, IMG_H);
            *(float4*)(lcol + r * XS_STRIDE) = *(const float4*)(gsrc + (size_t)gr * IMG_W);
        }
    }
    // ---- Stage the two halo columns (132 scalar elements)
    if (tid < 2 * XS_ROWS) {
        int r    = tid >> 1;
        int side = tid & 1;                       // 0 = left halo, 1 = right halo
        int gr = reflect101(row0 - 1 + r, IMG_H);
        int gc = reflect101(col0 - 1 + side * 129, IMG_W);
        xs[r * XS_STRIDE + 3 + side * 129] = src[gr * IMG_W + gc];
    }
    __syncthreads();

    // ---- Wave-private geometry ----
    const int lane = tid & 31;
    const int wv   = tid >> 5;       // 0..7: 16-col strip within the block
    const int sc   = wv * 16;        // image-col base of this strip
    const int m    = lane & 15;      // A-matrix row M / D-matrix col N for this lane
    const int kh   = lane >> 4;      // K-half: A/B VGPR t holds K = 2*kh + t

    // ---- Horizontal 1-2-1 band matrix B (20x16 in 5 K-chunks), built ONCE per wave.
    // B[e][n] = 0.25 (e==n or e==n+2), 0.5 (e==n+1), else 0; rows 18,19 are zero pad.
    // asm pin makes the values opaque -> no rematerialization inside the tile loop.
    v2f bmat[5];
    #pragma unroll
    for (int kk = 0; kk < 5; ++kk) {
        #pragma unroll
        for (int t = 0; t < 2; ++t) {
            int e = 4 * kk + 2 * kh + t;
            float bw = 0.0f;
            if (e == m || e == m + 2) bw = 0.25f;
            if (e == m + 1)           bw = 0.5f;
            asm("" : "+v"(bw));
            bmat[kk][t] = bw;
        }
    }

    const int gc = col0 + sc + m;    // output column this lane stores

    // ---- 4 stacked 16x16 output tiles per wave ----
    #pragma unroll
    for (int rt = 0; rt < 4; ++rt) {
        // Opaque per-tile base: all 30 LDS reads below become small constant
        // offsets from ONE VGPR base (rows m,m+1 fit ds_*_2addr's 8-bit*4B
        // offsets; row m+2 fits the 16-bit offset of plain ds_load_b32) instead
        // of per-load v_add address materialization.
        int off = (16 * rt + m) * XS_STRIDE + sc + 3;
        asm("" : "+v"(off));
        const float* prow = &xs[off];

        v8f acc = {};
        #pragma unroll
        for (int kk = 0; kk < 5; ++kk) {
            // A chunk: vertical 1-2-1 on the fly. Lane layout: M = lane&15,
            // VGPR t holds K = 2*kh + t (extended column ec). For ec=18,19 the
            // matching B rows are zero, so the A value is a don't-care: clamp
            // the read instead of masking -> branch-free, EXEC stays all-ones.
            v2f a;
            #pragma unroll
            for (int t = 0; t < 2; ++t) {
                int ec = 4 * kk + 2 * kh + t;
                if (ec > 17) ec = 17;
                const float* p = prow + ec;
                a[t] = 0.25f * (p[0] + p[2 * XS_STRIDE]) + 0.5f * p[XS_STRIDE];
            }
            // (neg_a, A, neg_b, B, c_mod, C, reuse_a, reuse_b)
            acc = __builtin_amdgcn_wmma_f32_16x16x4_f32(false, a, false, bmat[kk],
                                                        (short)0, acc, false, false);
        }
        // C/D layout: VGPR v -> M = v + 8*kh, N = lane&15. Output is write-once:
        // stream it past the caches, keep L2 for input halo reuse.
        #pragma unroll
        for (int v = 0; v < 8; ++v) {
            int gr = row0 + 16 * rt + v + 8 * kh;
            __builtin_nontemporal_store(acc[v], &dst[gr * IMG_W + gc]);
        }
    }
}

extern "C" void kernel_launch(void* const* d_in, const int* in_sizes, int n_in,
                              void* d_out, int out_size, void* d_ws, size_t ws_size,
                              hipStream_t stream) {
    const float* x = (const float*)d_in[0];
    float* out = (float*)d_out;
    dim3 grid(IMG_W / BLK_W, IMG_H / BLK_H, 64 * 3);   // 4 x 8 x 192 blocks
    gauss3x3_wmma_f32<<<grid, dim3(256), 0, stream>>>(x, out);
}